// DynamicReceptiveEncoder_90503550861341
// MI455X (gfx1250) — compile-verified
//
#include <hip/hip_runtime.h>
#include <stdint.h>

typedef float v2f __attribute__((ext_vector_type(2)));
typedef float v8f __attribute__((ext_vector_type(8)));

#define B_  32
#define W_  512
#define F_  25
#define C_  64
#define LDS_COLS 520   // col c <-> w = c-3 (zero pad cols 0..2, 515..519)
#define LDS_ROWS 9     // row r <-> frow = fa-3+r (rows outside [0,25) stay zero)

// Fused: 3x3 conv + 7x7 conv (f32 WMMA GEMM) + temporal delta + 4-way LIF scan.
// Block = 128 threads (4 waves). Block handles (batch b, f-pair fa/fa+1, all w).
// Wave g handles channels [16g, 16g+16). Tile GEMM: M = 2 f-rows x 8 w, N = 16 ch,
// K = im2col taps (52-padded for 7x7, 12-padded for 3x3).
__global__ __launch_bounds__(128)
void dre_fused_kernel(const float* __restrict__ in,   // [B, W, F]
                      const float* __restrict__ w3,   // [C,1,3,3]
                      const float* __restrict__ b3,   // [C]
                      const float* __restrict__ w7,   // [C,1,7,7]
                      const float* __restrict__ b7,   // [C]
                      float* __restrict__ out)        // [B, C, F, W]
{
    __shared__ float ximg[LDS_ROWS * LDS_COLS];

    const int tid  = threadIdx.x;
    const int lane = tid & 31;
    const int wave = tid >> 5;          // channel group 0..3
    const int half = lane >> 4;         // K-pair selector (A/B operands)
    const int mA   = lane & 15;         // A-matrix row this lane feeds
    const int blk  = blockIdx.x;
    const int b    = blk / 13;
    const int fa   = (blk % 13) * 2;    // f-pair (fa, fa+1)
    const int c    = wave * 16 + (lane & 15);  // B/D column: this lane's channel
    const int fv   = fa + half;                // D row group: this lane's f row
    const int dfA  = mA >> 3;           // A-row f offset (0 -> fa, 1 -> fa+1)
    const int woffA = mA & 7;           // A-row w offset within chunk

    // ---- zero the padded LDS image ----
    for (int i = tid; i < LDS_ROWS * LDS_COLS; i += 128) ximg[i] = 0.0f;
    __syncthreads();

    // ---- async-stage rows fa-3..fa+4 (strided gather from [B,W,F]) ----
    {
        const uint32_t lds_base = (uint32_t)(uintptr_t)(&ximg[0]); // low 32b = LDS offset
        for (int i = tid; i < 8 * W_; i += 128) {
            const int r = i >> 9;             // 0..7
            const int w = i & (W_ - 1);
            const int frow = fa - 3 + r;
            if (frow >= 0 && frow < F_) {
                const uint32_t loff = lds_base + (uint32_t)((r * LDS_COLS + 3 + w) * 4);
                const float* gaddr = in + ((size_t)(b * W_ + w) * F_ + frow);
                asm volatile("global_load_async_to_lds_b32 %0, %1, off"
                             :: "v"(loff), "v"(gaddr) : "memory");
            }
        }
        asm volatile("s_wait_asynccnt 0x0" ::: "memory");
    }
    __syncthreads();

    // ---- preload weights into WMMA B layout (K = 4t + 2*half + j, N = c) ----
    v2f Bw7[13];
#pragma unroll
    for (int t = 0; t < 13; ++t) {
#pragma unroll
        for (int j = 0; j < 2; ++j) {
            const int k = 4 * t + 2 * half + j;
            Bw7[t][j] = (k < 49) ? w7[c * 49 + k] : 0.0f;
        }
    }
    v2f Bw3[3];
#pragma unroll
    for (int t = 0; t < 3; ++t) {
#pragma unroll
        for (int j = 0; j < 2; ++j) {
            const int k = 4 * t + 2 * half + j;
            Bw3[t][j] = (k < 9) ? w3[c * 9 + k] : 0.0f;
        }
    }
    const float bias3 = b3[c];
    const float bias7 = b7[c];

    // LIF state (per lane: one (c, f) pair), temporal-delta history
    float v1 = 0.f, v2 = 0.f, v3 = 0.f, v4 = 0.f;
    float prev3 = 0.f, prev7 = 0.f, gate = 0.f;

    const bool  do_store = (fv < F_);
    float* outp = do_store ? (out + ((size_t)(b * C_ + c) * F_ + fv) * W_) : nullptr;

    const float dec1 = 1.0f - 1.0f / 20.0f;
    const float dec2 = 1.0f - 1.0f / 50.0f;
    const float dec3 = 1.0f - 1.0f / 2.0f;
    const float dec4 = 1.0f - 1.0f / 0.91f;

    for (int w0 = 0; w0 < W_; w0 += 8) {
        v8f acc7 = {0.f,0.f,0.f,0.f,0.f,0.f,0.f,0.f};
        v8f acc3 = {0.f,0.f,0.f,0.f,0.f,0.f,0.f,0.f};

        // 7x7 conv: D[m][n] += patch[m][k] * W7[k][n], K padded 49 -> 52
#pragma unroll
        for (int t = 0; t < 13; ++t) {
            v2f A;
#pragma unroll
            for (int j = 0; j < 2; ++j) {
                const int k  = 4 * t + 2 * half + j;
                const int kk = (k < 49) ? k : 0;       // padded K: weight is 0
                const int kf = kk / 7, kw = kk % 7;
                A[j] = ximg[(dfA + kf) * LDS_COLS + (w0 + woffA + kw)];
            }
            acc7 = __builtin_amdgcn_wmma_f32_16x16x4_f32(
                       false, A, false, Bw7[t], (short)0, acc7, false, false);
        }
        // 3x3 conv: K padded 9 -> 12 (rows +2, cols +2 to re-center pad=1 in pad=3 image)
#pragma unroll
        for (int t = 0; t < 3; ++t) {
            v2f A;
#pragma unroll
            for (int j = 0; j < 2; ++j) {
                const int k  = 4 * t + 2 * half + j;
                const int kk = (k < 9) ? k : 0;
                const int kf = kk / 3, kw = kk % 3;
                A[j] = ximg[(dfA + kf + 2) * LDS_COLS + (w0 + woffA + kw + 2)];
            }
            acc3 = __builtin_amdgcn_wmma_f32_16x16x4_f32(
                       false, A, false, Bw3[t], (short)0, acc3, false, false);
        }

        // ---- sequential LIF scan over the 8 w-steps held in this lane's D VGPRs ----
        float sbuf[8];
#pragma unroll
        for (int r = 0; r < 8; ++r) {
            const float a3v = acc3[r] + bias3;
            const float a7v = acc7[r] + bias7;
            const float d3v = gate * fabsf(a3v - prev3);
            const float d7v = gate * fabsf(a7v - prev7);
            prev3 = a3v; prev7 = a7v; gate = 1.0f;

            v1 = v1 * dec1 + a3v;
            const float s1 = (v1 >= 1.0f) ? 1.0f : 0.0f;  v1 = (v1 >= 1.0f) ? 0.0f : v1;
            v2 = v2 * dec2 + a7v;
            const float s2 = (v2 >= 1.0f) ? 1.0f : 0.0f;  v2 = (v2 >= 1.0f) ? 0.0f : v2;
            v3 = v3 * dec3 + d3v;
            const float s3 = (v3 >= 0.8f) ? 1.0f : 0.0f;  v3 = (v3 >= 0.8f) ? 0.0f : v3;
            v4 = v4 * dec4 + d7v;
            const float s4 = (v4 >= 0.8f) ? 1.0f : 0.0f;  v4 = (v4 >= 0.8f) ? 0.0f : v4;
            sbuf[r] = s1 + s2 + s3 + s4;
        }
        if (do_store) {
            float4 lo = {sbuf[0], sbuf[1], sbuf[2], sbuf[3]};
            float4 hi = {sbuf[4], sbuf[5], sbuf[6], sbuf[7]};
            *(float4*)(outp + w0)     = lo;
            *(float4*)(outp + w0 + 4) = hi;
        }
    }
}

extern "C" void kernel_launch(void* const* d_in, const int* in_sizes, int n_in,
                              void* d_out, int out_size, void* d_ws, size_t ws_size,
                              hipStream_t stream) {
    (void)in_sizes; (void)n_in; (void)out_size; (void)d_ws; (void)ws_size;
    const float* in = (const float*)d_in[0];
    const float* w3 = (const float*)d_in[1];
    const float* b3 = (const float*)d_in[2];
    const float* w7 = (const float*)d_in[3];
    const float* b7 = (const float*)d_in[4];
    float* outp = (float*)d_out;

    dim3 grid(B_ * 13);   // 32 batches x 13 f-pairs
    dim3 block(128);      // 4 waves = 4 channel groups
    hipLaunchKernelGGL(dre_fused_kernel, grid, block, 0, stream,
                       in, w3, b3, w7, b7, outp);
}